// MobileMQA_5772436046353
// MI455X (gfx1250) — compile-verified
//
#include <hip/hip_runtime.h>
#include <hip/hip_bf16.h>
#include <math.h>

typedef float v2f __attribute__((ext_vector_type(2)));
typedef float v8f __attribute__((ext_vector_type(8)));
typedef __attribute__((address_space(3))) float lds_f;

#define DIM 512
#define HEADS 8
#define HD 64
#define L 1024          // 32*32 per batch
#define KVM 2048        // 16*16*8 per batch
#define SCALE 0.125f    // 64^-0.5
#define KC 32           // GEMM k-chunk staged in LDS

#define USE_ASYNC_STAGE 1

// ---------------------------------------------------------------------------
// fp32 GEMM + bias using V_WMMA_F32_16X16X4_F32.
// Block (256 thr / 8 waves) computes a 128(M) x 16(N) macro-tile; the shared
// B chunk (KC x 16) is staged into LDS once per k-chunk via async-to-LDS
// copies (ASYNCcnt) and consumed by all 8 waves as ds_load_b64 B-fragments.
// LDS layout transposed with stride 34: bs[n*34 + kk] -> lane r reads the
// aligned b64 pair {B[kk][n0+r], B[kk+1][n0+r]} conflict-free.
// A-frag: lane r holds row m0+r, K pair {2*half, 2*half+1} (b64 from global).
// D: acc[i] = C[m0 + i + 8*half][n0 + r].
// ---------------------------------------------------------------------------
__global__ __launch_bounds__(256) void gemm_bias_wmma(
    const float* __restrict__ A, const float* __restrict__ W,
    const float* __restrict__ bias, float* __restrict__ C,
    int M, int N, int K)
{
  const int tid  = threadIdx.x;
  const int lane = tid & 31;
  const int wave = tid >> 5;
  const int tilesN = N >> 4;
  const int tn = blockIdx.x % tilesN;
  const int bm = blockIdx.x / tilesN;
  const int half = lane >> 4;
  const int r = lane & 15;
  const int m0 = bm * 128 + wave * 16;
  const int n0 = tn << 4;

  __shared__ float bs[16 * 34];   // B chunk, transposed, stride 34 (even)

  const int sn = tid & 15;        // staged column
  const int sk = tid >> 4;        // staged k-row; this thread does sk, sk+16

  v8f acc = {};
  const float* arow = A + (size_t)(m0 + r) * K + 2 * half;

  for (int k0 = 0; k0 < K; k0 += KC) {
    // ---- stage B chunk [k0..k0+KC) x [n0..n0+16) into LDS ----
    const float* g0 = W + (size_t)(k0 + sk)      * N + n0 + sn;
    const float* g1 = W + (size_t)(k0 + sk + 16) * N + n0 + sn;
#if USE_ASYNC_STAGE
    {
      unsigned l0 = (unsigned)(uintptr_t)(lds_f*)&bs[sn * 34 + sk];
      unsigned l1 = l0 + 16u * 4u;
      asm volatile("global_load_async_to_lds_b32 %0, %1, off"
                   :: "v"(l0), "v"((unsigned long long)(uintptr_t)g0)
                   : "memory");
      asm volatile("global_load_async_to_lds_b32 %0, %1, off"
                   :: "v"(l1), "v"((unsigned long long)(uintptr_t)g1)
                   : "memory");
      asm volatile("s_wait_asynccnt 0x0" ::: "memory");
    }
#else
    bs[sn * 34 + sk]      = *g0;
    bs[sn * 34 + sk + 16] = *g1;
#endif
    __syncthreads();

    // prefetch next A chunk for this lane's row (speculative, dropped if OOB)
    __builtin_prefetch(arow + k0 + KC, 0, 0);

    // ---- KC/4 = 8 WMMA k-steps ----
#pragma unroll
    for (int ks = 0; ks < KC; ks += 4) {
      v2f a = *(const v2f*)(arow + k0 + ks);
      v2f b = *(const v2f*)&bs[r * 34 + ks + 2 * half];
      acc = __builtin_amdgcn_wmma_f32_16x16x4_f32(false, a, false, b,
                                                  (short)0, acc, false, false);
    }
    __syncthreads();
  }

  const float bv = bias[n0 + r];
  float* crow = C + (size_t)m0 * N + n0 + r;
#pragma unroll
  for (int i = 0; i < 8; ++i)
    crow[(size_t)(i + 8 * half) * N] = acc[i] + bv;
}

// ---------------------------------------------------------------------------
// DepthwiseConv2D 3x3 stride 2, SAME padding (pad_lo=0, pad_hi=1), NHWC.
// in: [2,32,32,1024]  kern: [3,3,1,1024]  out: [2,16,16,1024]
// ---------------------------------------------------------------------------
__global__ __launch_bounds__(256) void dwconv_s2(
    const float* __restrict__ in, const float* __restrict__ kern,
    const float* __restrict__ bias, float* __restrict__ out)
{
  int idx = blockIdx.x * 256 + threadIdx.x;     // 2*16*16*1024 total
  int c  = idx & 1023;
  int t  = idx >> 10;
  int ow = t & 15; t >>= 4;
  int oh = t & 15; t >>= 4;
  int b  = t;
  float s = bias[c];
#pragma unroll
  for (int kh = 0; kh < 3; ++kh) {
    int ih = oh * 2 + kh;
    if (ih >= 32) continue;
#pragma unroll
    for (int kw = 0; kw < 3; ++kw) {
      int iw = ow * 2 + kw;
      if (iw >= 32) continue;
      s += in[(((size_t)b * 32 + ih) * 32 + iw) * 1024 + c] *
           kern[(kh * 3 + kw) * 1024 + c];
    }
  }
  out[idx] = s;
}

// ---------------------------------------------------------------------------
// MQA attention, flash-style online softmax, fp32 WMMA.
// One wave per (batch, head, 16-row q tile). Q A-frags persist in registers;
// K/V gathered straight from the folded kv tensor (channel-fold reshape:
// m = s*8 + n2, k channel = n2*64+d, v channel = 512 + n2*64+d).
// ---------------------------------------------------------------------------
__global__ __launch_bounds__(32) void attn_wmma(
    const float* __restrict__ q,    // [2, 1024, 512]
    const float* __restrict__ kvd,  // [2, 256, 1024]
    float* __restrict__ o)          // [2, 1024, 512]
{
  int t  = blockIdx.x;              // 2*8*64 blocks
  int qt = t & 63;  t >>= 6;
  int h  = t & 7;   t >>= 3;
  int b  = t;
  const int lane = threadIdx.x;
  const int half = lane >> 4;
  const int r = lane & 15;
  const int qrow0 = qt << 4;

  // Preload Q A-fragments: qa[j] covers K = 4j..4j+3
  const float* qbase = q + ((size_t)b * L + qrow0) * DIM + h * HD;
  v2f qa[16];
#pragma unroll
  for (int j = 0; j < 16; ++j)
    qa[j] = *(const v2f*)(qbase + (size_t)r * DIM + 4 * j + 2 * half);

  const float* kbase = kvd + (size_t)b * 256 * 1024;

  float mx[8], ls[8];
#pragma unroll
  for (int i = 0; i < 8; ++i) { mx[i] = -3.0e38f; ls[i] = 0.0f; }
  v8f oacc[4] = {};               // oacc[tt][i] = O[i+8*half][tt*16 + r]

  __shared__ float plds[16][17];  // P tile relay (padded vs bank conflicts)

  for (int m0 = 0; m0 < KVM; m0 += 16) {
    // ---- S = Q @ K^T ----
    v8f s = {};
    const int m = m0 + r;          // this lane's K/V row (column of S)
    const float* krow = kbase + (size_t)(m >> 3) * 1024 + (m & 7) * HD;
#pragma unroll
    for (int j = 0; j < 16; ++j) {
      v2f kb = *(const v2f*)(krow + 4 * j + 2 * half);
      s = __builtin_amdgcn_wmma_f32_16x16x4_f32(false, qa[j], false, kb,
                                                (short)0, s, false, false);
    }
    // ---- online softmax over the 16 columns held across the half ----
    float p[8], alpha[8];
#pragma unroll
    for (int i = 0; i < 8; ++i) {
      float sv = s[i] * SCALE;
      float m_ = sv;
      m_ = fmaxf(m_, __shfl_xor(m_, 1, 16));
      m_ = fmaxf(m_, __shfl_xor(m_, 2, 16));
      m_ = fmaxf(m_, __shfl_xor(m_, 4, 16));
      m_ = fmaxf(m_, __shfl_xor(m_, 8, 16));
      float newm = fmaxf(mx[i], m_);
      alpha[i] = __expf(mx[i] - newm);
      mx[i] = newm;
      float pv = __expf(sv - newm);
      p[i] = pv;
      float rs = pv;
      rs += __shfl_xor(rs, 1, 16);
      rs += __shfl_xor(rs, 2, 16);
      rs += __shfl_xor(rs, 4, 16);
      rs += __shfl_xor(rs, 8, 16);
      ls[i] = ls[i] * alpha[i] + rs;
    }
#pragma unroll
    for (int tt = 0; tt < 4; ++tt)
#pragma unroll
      for (int i = 0; i < 8; ++i)
        oacc[tt][i] *= alpha[i];

    // relay P (D layout) -> A layout through LDS
#pragma unroll
    for (int i = 0; i < 8; ++i)
      plds[i + 8 * half][r] = p[i];
    __syncthreads();

    // ---- O += P @ V ----
#pragma unroll
    for (int j = 0; j < 4; ++j) {
      const int kk = 4 * j + 2 * half;
      v2f pa;
      pa.x = plds[r][kk];
      pa.y = plds[r][kk + 1];
      const int m1 = m0 + kk;
      const float* v0 = kbase + (size_t)(m1 >> 3) * 1024 + (m1 & 7) * HD + DIM;
      const float* v1 = kbase + (size_t)((m1 + 1) >> 3) * 1024 +
                        ((m1 + 1) & 7) * HD + DIM;
#pragma unroll
      for (int tt = 0; tt < 4; ++tt) {
        const int d = tt * 16 + r;
        v2f vb;
        vb.x = v0[d];
        vb.y = v1[d];
        oacc[tt] = __builtin_amdgcn_wmma_f32_16x16x4_f32(false, pa, false, vb,
                                                         (short)0, oacc[tt],
                                                         false, false);
      }
    }
    __syncthreads();
  }

  // epilogue: divide by row sums, scatter to [b, l, h*64 + d]
  float* obase = o + ((size_t)b * L + qrow0) * DIM + h * HD;
#pragma unroll
  for (int tt = 0; tt < 4; ++tt)
#pragma unroll
    for (int i = 0; i < 8; ++i)
      obase[(size_t)(i + 8 * half) * DIM + tt * 16 + r] = oacc[tt][i] / ls[i];
}

// ---------------------------------------------------------------------------
extern "C" void kernel_launch(void* const* d_in, const int* in_sizes, int n_in,
                              void* d_out, int out_size, void* d_ws, size_t ws_size,
                              hipStream_t stream) {
  const float* x    = (const float*)d_in[0];  // [2,32,32,512]
  const float* wq   = (const float*)d_in[1];  // [512,512]
  const float* bq   = (const float*)d_in[2];  // [512]
  const float* wkv  = (const float*)d_in[3];  // [512,1024]
  const float* bkv  = (const float*)d_in[4];  // [1024]
  const float* dwk  = (const float*)d_in[5];  // [3,3,1,1024]
  const float* dwb  = (const float*)d_in[6];  // [1024]
  const float* wo   = (const float*)d_in[7];  // [512,512]
  const float* bo   = (const float*)d_in[8];  // [512]
  float* out = (float*)d_out;                 // [2,32,32,512]

  float* q_ws   = (float*)d_ws;               // 2048*512
  float* kv_ws  = q_ws   + 2048 * 512;        // 2048*1024
  float* kvd_ws = kv_ws  + 2048 * 1024;       // 2*256*1024
  float* ao_ws  = kvd_ws + 512 * 1024;        // 2048*512

  // q = x@wq + bq : (2048/128)*(512/16) = 512 blocks
  gemm_bias_wmma<<<512, 256, 0, stream>>>(x, wq, bq, q_ws, 2048, 512, 512);
  // kv = x@wkv + bkv : (2048/128)*(1024/16) = 1024 blocks
  gemm_bias_wmma<<<1024, 256, 0, stream>>>(x, wkv, bkv, kv_ws, 2048, 1024, 512);
  // depthwise 3x3 s2 + bias : 2*16*16*1024 = 524288 elems
  dwconv_s2<<<2048, 256, 0, stream>>>(kv_ws, dwk, dwb, kvd_ws);
  // attention: 2 batches * 8 heads * 64 q-tiles = 1024 waves
  attn_wmma<<<1024, 32, 0, stream>>>(q_ws, kvd_ws, ao_ws);
  // out = attn_out@wo + bo
  gemm_bias_wmma<<<512, 256, 0, stream>>>(ao_ws, wo, bo, out, 2048, 512, 512);
}